// LookUpDurationModel_15367392985794
// MI455X (gfx1250) — compile-verified
//
#include <hip/hip_runtime.h>
#include <hip/hip_bf16.h>
#include <climits>

// Workload: B=1024, S=4096, P=512 duration-model scatter/avg + per-row gather/rescale.
// Memory-bound (~64 MiB total traffic -> ~3us at 23.3 TB/s). No matmul -> no WMMA;
// CDNA5-specific paths used: LDS-privatized ds_add_f32 histogram atomics and
// global_load_async_to_lds_b128 / s_wait_asynccnt row staging.

#define NPHONE 512
#define SEQ    4096

// ---- workspace layout (in floats) ----
// [0,512)        sum_acc
// [512,1024)     cnt_acc
// [1024]         has26 flag (int)
// [1028..1029]   scal: dn_i, num
// [1040,1552)    dur_table (int)

__global__ void k_init(float* __restrict__ ws) {
  int i = blockIdx.x * blockDim.x + threadIdx.x;
  if (i < 1040) ws[i] = 0.0f;
}

__global__ void __launch_bounds__(256)
k_scatter(const int* __restrict__ idx, const float* __restrict__ ds,
          float* __restrict__ sum_acc, float* __restrict__ cnt_acc,
          int* __restrict__ has26, int N) {
  __shared__ float lsum[NPHONE];
  __shared__ float lcnt[NPHONE];
  __shared__ int   blk26;
  for (int p = threadIdx.x; p < NPHONE; p += blockDim.x) { lsum[p] = 0.0f; lcnt[p] = 0.0f; }
  if (threadIdx.x == 0) blk26 = 0;
  __syncthreads();

  int l26 = 0;
  const int tid    = blockIdx.x * blockDim.x + threadIdx.x;
  const int stride = gridDim.x * blockDim.x;

  // vectorized b128 streaming: each thread handles 4 consecutive elements/step
  const int N4 = N >> 2;
  const int4*   idx4 = (const int4*)idx;
  const float4* ds4  = (const float4*)ds;
  for (int i = tid; i < N4; i += stride) {
    int4   p = idx4[i];
    float4 d = ds4[i];
    atomicAdd(&lsum[p.x], d.x); atomicAdd(&lcnt[p.x], 1.0f);
    atomicAdd(&lsum[p.y], d.y); atomicAdd(&lcnt[p.y], 1.0f);
    atomicAdd(&lsum[p.z], d.z); atomicAdd(&lcnt[p.z], 1.0f);
    atomicAdd(&lsum[p.w], d.w); atomicAdd(&lcnt[p.w], 1.0f);
    l26 |= ((p.x > 1) && (p.x < 7)) | ((p.y > 1) && (p.y < 7)) |
           ((p.z > 1) && (p.z < 7)) | ((p.w > 1) && (p.w < 7));
  }
  // scalar tail (N not multiple of 4)
  for (int i = (N4 << 2) + tid; i < N; i += stride) {
    int   p = idx[i];
    float d = ds[i];
    atomicAdd(&lsum[p], d);
    atomicAdd(&lcnt[p], 1.0f);
    l26 |= (p > 1) && (p < 7);
  }

  if (l26) atomicOr(&blk26, 1);  // shared, cheap
  __syncthreads();

  for (int p = threadIdx.x; p < NPHONE; p += blockDim.x) {
    float c = lcnt[p];
    if (c > 0.0f) {
      atomicAdd(&sum_acc[p], lsum[p]);   // one L2 atomic per bin per block
      atomicAdd(&cnt_acc[p], c);
    }
  }
  if (threadIdx.x == 0 && blk26) atomicOr(has26, 1);
}

__global__ void __launch_bounds__(NPHONE)
k_finalize(const float* __restrict__ sum0, const float* __restrict__ cnt0,
           const float* __restrict__ dur_prev, const float* __restrict__ rv,
           const float* __restrict__ dn_in, const float* __restrict__ sum_acc,
           const float* __restrict__ cnt_acc, const int* __restrict__ has26,
           float* __restrict__ out_tab, float* __restrict__ out_dn,
           int* __restrict__ dur_table, float* __restrict__ scal) {
  __shared__ float sh_s[NPHONE];
  __shared__ float sh_c[NPHONE];
  const int p = threadIdx.x;
  float s = sum0[p] + sum_acc[p];
  float c = cnt0[p] + cnt_acc[p];
  bool touched = cnt_acc[p] > 0.0f;
  float dnv = touched ? (s / fmaxf(c, 1.0f)) : dur_prev[p];
  out_tab[p]   = dnv;                 // duration_new output
  dur_table[p] = (int)truncf(dnv);    // int lookup table for eval phase
  sh_s[p] = s; sh_c[p] = c;
  __syncthreads();
  if (p == 0) {
    float a = 0.0f, b = 0.0f;
    for (int q = 2; q < 7; ++q) { a += sh_s[q]; b += sh_c[q]; }
    float dn_new = (*has26) ? (a / b) : dn_in[0];
    out_dn[0] = dn_new;               // dn_new output
    float dni = truncf(dn_new);
    scal[0] = dni;                              // dn_i
    scal[1] = dni - truncf(rv[0] * dn_new);     // num = int(dn) - int(rv*dn)
  }
}

__global__ void __launch_bounds__(256)
k_eval(const int* __restrict__ idx, const int* __restrict__ dur_table,
       const float* __restrict__ scal, const int* __restrict__ pad_idx_p,
       float* __restrict__ out) {
  __shared__ __align__(16) int sidx[SEQ];   // 16 KB idx row staged in LDS
  __shared__ int tab[NPHONE];
  __shared__ int s_min, s_max, s_den;
  const int b = blockIdx.x;
  const int t = threadIdx.x;
  const int* row = idx + (size_t)b * SEQ;

#if defined(__AMDGCN__)
  {
    // Async global->LDS copy of the row: 4 x b128 per lane, tracked by ASYNCcnt.
    unsigned lds0 = (unsigned)(size_t)&sidx[0];
#pragma unroll
    for (int c = 0; c < 4; ++c) {
      unsigned loff = lds0 + (unsigned)(t * 16 + c * 4096);
      const void* g = (const void*)(row + t * 4 + c * 1024);
      asm volatile("global_load_async_to_lds_b128 %0, %1, off"
                   :: "v"(loff), "v"(g) : "memory");
    }
  }
#else
  for (int j = t; j < SEQ; j += 256) sidx[j] = row[j];
#endif
  for (int p = t; p < NPHONE; p += blockDim.x) tab[p] = dur_table[p];
  if (t == 0) { s_min = SEQ; s_max = 0; s_den = 0; }
#if defined(__AMDGCN__)
  asm volatile("s_wait_asynccnt 0" ::: "memory");  // my wave's async loads done
#endif
  __syncthreads();                                 // all waves' data visible

  // pass 1: first padding position (j>=1) and max(dur[1:])
  const int padv = pad_idx_p[0];
  int lmin = SEQ, lmax = 0;
  for (int j = t; j < SEQ; j += 256) {
    int v = sidx[j];
    int d = tab[v];
    if (j >= 1) {
      if (v == padv && j < lmin) lmin = j;
      if (d > lmax) lmax = d;
    }
  }
  atomicMin(&s_min, lmin);
  atomicMax(&s_max, lmax);
  __syncthreads();

  const int n = (s_min < SEQ) ? s_min : 1;         // first pad pos, else 1

  // pass 2: denom = sum(dur[1:n])
  int lden = 0;
  for (int j = t; j < n; j += 256)
    if (j >= 1) lden += tab[sidx[j]];
  if (lden) atomicAdd(&s_den, lden);
  __syncthreads();

  const float dn_i = scal[0];
  const float num  = scal[1];
  float rc;
  if (n == 1) rc = 1.0f;
  else {
    float den = (s_den > 0) ? (float)s_den : 1.0f;
    rc = fminf(1.0f, num / den);
  }
  const int delta = (s_max > 1) ? s_max : 1;
  const int d0    = max(1, (int)dn_i - delta);

  // pass 3: write rescaled row
  float* orow = out + (size_t)b * SEQ;
  for (int j = t; j < SEQ; j += 256) {
    int d = tab[sidx[j]];
    int ov;
    if (j == 0)      ov = d0;
    else if (j < n) { int r = (int)(rc * (float)d); ov = (r > 1) ? r : 1; }
    else             ov = d;
    orow[j] = (float)ov;
  }
}

extern "C" void kernel_launch(void* const* d_in, const int* in_sizes, int n_in,
                              void* d_out, int out_size, void* d_ws, size_t ws_size,
                              hipStream_t stream) {
  (void)n_in; (void)out_size; (void)ws_size;
  const int*   idx   = (const int*)d_in[0];
  const float* ds    = (const float*)d_in[1];
  const float* sum0  = (const float*)d_in[2];
  const float* cnt0  = (const float*)d_in[3];
  const float* dur0  = (const float*)d_in[4];
  const float* rv    = (const float*)d_in[5];
  const float* dn    = (const float*)d_in[6];
  const int*   padix = (const int*)d_in[7];

  const int N = in_sizes[0];      // B * S
  const int B = N / SEQ;

  float* ws       = (float*)d_ws;
  float* sum_acc  = ws;
  float* cnt_acc  = ws + NPHONE;
  int*   has26    = (int*)(ws + 2 * NPHONE);
  float* scal     = ws + 2 * NPHONE + 4;
  int*   durtab   = (int*)(ws + 2 * NPHONE + 16);

  float* out      = (float*)d_out;
  float* out_tab  = out + (size_t)N;        // duration_new [P]
  float* out_dn   = out_tab + NPHONE;       // dn_new scalar

  k_init<<<(1040 + 255) / 256, 256, 0, stream>>>(ws);
  k_scatter<<<512, 256, 0, stream>>>(idx, ds, sum_acc, cnt_acc, has26, N);
  k_finalize<<<1, NPHONE, 0, stream>>>(sum0, cnt0, dur0, rv, dn,
                                       sum_acc, cnt_acc, has26,
                                       out_tab, out_dn, durtab, scal);
  k_eval<<<B, 256, 0, stream>>>(idx, durtab, scal, padix, out);
}